// HybridQLSTM_65481071404001
// MI455X (gfx1250) — compile-verified
//
#include <hip/hip_runtime.h>
#include <hip/hip_bf16.h>

// ---------------------------------------------------------------------------
// HybridQLSTM for MI455X (gfx1250)
//
// Decomposition:
//  1) prep_kernel     : repack gate weights -> WxT(32x16 k-major), WhA(16x4),
//                       bA(16), thA(16), wr(4), br  into first 4KB of d_ws.
//  2) gemm_pre_kernel : pre[t,b,g] = x[t,b,:] @ WxT[:,g] + bA[g]  for all T*B
//                       rows at once, using V_WMMA_F32_16X16X4_F32 (full f32).
//  3) step_kernel x256: serial scan. Closed-form 4-qubit layer:
//                       RX(a)RX(th)=RX(a+th) on |0>, product state, CNOT ring
//                       is a basis permutation => expectations are signed
//                       sums of factorized probabilities (4 cos per gate).
//     h/c live in their final d_out slots and act as the running state.
// ---------------------------------------------------------------------------

typedef __attribute__((ext_vector_type(2))) float v2f;
typedef __attribute__((ext_vector_type(8))) float v8f;

#define TT 256
#define BB 8192

// ws constant block layout (floats):
//  [0..511]   WxT   : WxT[k*16+g] = W_gate(g)[q, k]   (k-major, 32x16)
//  [512..575] WhA   : WhA[g*4+j]  = W_gate(g)[q, 32+j]
//  [576..591] bA
//  [592..607] thA
//  [608..611] wr
//  [612]      br
#define CST_WXT 0
#define CST_WH  512
#define CST_B   576
#define CST_TH  592
#define CST_WR  608
#define CST_BR  612

__global__ void prep_kernel(const float* __restrict__ Wf, const float* __restrict__ bf,
                            const float* __restrict__ Wi, const float* __restrict__ bi,
                            const float* __restrict__ Wu, const float* __restrict__ bu,
                            const float* __restrict__ Wo, const float* __restrict__ bo,
                            const float* __restrict__ thf, const float* __restrict__ thi,
                            const float* __restrict__ thu, const float* __restrict__ tho,
                            const float* __restrict__ Wr, const float* __restrict__ br,
                            float* __restrict__ cst) {
  const float* Wg[4] = {Wf, Wi, Wu, Wo};
  const float* bg[4] = {bf, bi, bu, bo};
  const float* tg[4] = {thf, thi, thu, tho};
  int t = threadIdx.x;
  for (int idx = t; idx < 512; idx += blockDim.x) {
    int k = idx >> 4, g = idx & 15;
    cst[CST_WXT + idx] = Wg[g >> 2][(g & 3) * 36 + k];
  }
  if (t < 64) {
    int g = t >> 2, j = t & 3;
    cst[CST_WH + g * 4 + j] = Wg[g >> 2][(g & 3) * 36 + 32 + j];
  }
  if (t < 16) {
    cst[CST_B + t]  = bg[t >> 2][t & 3];
    cst[CST_TH + t] = tg[t >> 2][t & 3];
  }
  if (t < 4) cst[CST_WR + t] = Wr[t];
  if (t == 0) cst[CST_BR] = br[0];
}

// Zero h/c running state (lives at tail of d_out -> becomes hT/cT).
__global__ void init_state(float* __restrict__ out) {
  int i = blockIdx.x * blockDim.x + threadIdx.x;
  float* hb = out + (size_t)TT * BB * 5;  // after stacked (T*B*4) + regress (T*B)
  if (i < 2 * BB * 4) hb[i] = 0.0f;
}

// pre = X @ WxT + bias over all T*B rows. One wave per 16-row tile.
// A (16x4 f32): lanes 0-15 M=lane K={0,1}; lanes 16-31 M=lane-16 K={2,3}.
// B (4x16 f32): same striping with N in place of M.
// C (16x16 f32): vgpr j -> lanes 0-15 M=j, lanes 16-31 M=j+8, N=lane&15.
__global__ void __launch_bounds__(128)
gemm_pre_kernel(const float* __restrict__ X, const float* __restrict__ cst,
                float* __restrict__ pre, int numRows) {
  int wave = (int)((blockIdx.x * blockDim.x + threadIdx.x) >> 5);
  int lane = threadIdx.x & 31;
  int row0 = wave << 4;
  if (row0 >= numRows) return;                 // uniform per-wave: EXEC stays all-ones
  int M    = lane & 15;
  int koff = (lane < 16) ? 0 : 2;

  const float* xrow = X + (size_t)(row0 + M) * 32 + koff;
  const float* wcol = cst + CST_WXT + M;       // WxT[k*16 + M]

  v8f acc = {};
#pragma unroll
  for (int kk = 0; kk < 8; ++kk) {
    int k = kk * 4 + koff;
    v2f a, b;
    a.x = xrow[kk * 4 + 0];
    a.y = xrow[kk * 4 + 1];
    b.x = wcol[(k + 0) * 16];
    b.y = wcol[(k + 1) * 16];
    acc = __builtin_amdgcn_wmma_f32_16x16x4_f32(false, a, false, b, (short)0, acc,
                                                false, false);
  }

  float bias = cst[CST_B + M];
  int mbase  = (lane < 16) ? 0 : 8;
  float* orow = pre + (size_t)(row0 + mbase) * 16 + M;
#pragma unroll
  for (int j = 0; j < 8; ++j) orow[(size_t)j * 16] = acc[j] + bias;
}

__device__ __forceinline__ float sigmoidf_(float x) {
  return 1.0f / (1.0f + __expf(-x));
}

// One LSTM timestep for all B elements (1 thread = 1 batch element).
__global__ void __launch_bounds__(256)
step_kernel(const float* __restrict__ src, const float* __restrict__ cst,
            float* __restrict__ out, int t, int usePre) {
  int b = blockIdx.x * blockDim.x + threadIdx.x;
  if (b >= BB) return;

  float* stacked = out;                                  // (T,B,4)
  float* regress = out + (size_t)TT * BB * 4;            // (T,B)
  float* hbuf    = out + (size_t)TT * BB * 5;            // (B,4) running h -> hT
  float* cbuf    = hbuf + (size_t)BB * 4;                // (B,4) running c -> cT

  float h[4], c[4];
#pragma unroll
  for (int j = 0; j < 4; ++j) { h[j] = hbuf[b * 4 + j]; c[j] = cbuf[b * 4 + j]; }

  float ang[16];
  if (usePre) {
    const float* p = src + ((size_t)t * BB + b) * 16;
#pragma unroll
    for (int g = 0; g < 16; ++g) ang[g] = p[g];
  } else {
    const float* x = src + ((size_t)t * BB + b) * 32;
    float xv[32];
#pragma unroll
    for (int k = 0; k < 32; ++k) xv[k] = x[k];
#pragma unroll
    for (int g = 0; g < 16; ++g) {
      float a = cst[CST_B + g];
#pragma unroll
      for (int k = 0; k < 32; ++k) a += xv[k] * cst[CST_WXT + k * 16 + g];
      ang[g] = a;
    }
  }

  // recurrent part: ang += h @ WhA.T
#pragma unroll
  for (int g = 0; g < 16; ++g) {
    float a = ang[g];
#pragma unroll
    for (int j = 0; j < 4; ++j) a += h[j] * cst[CST_WH + g * 4 + j];
    ang[g] = a;
  }

  // Quantum layers, closed form.
  // RX(a)RX(th) = RX(a+th) applied to |0000> -> product state with
  // per-wire probabilities p0=(1+cosA)/2, p1=(1-cosA)/2. CNOT ring only
  // permutes basis states; <Z_w> = sum_i sign_w(ring(i)) * prod_w p(i).
  float gates[16];
#pragma unroll
  for (int gate = 0; gate < 4; ++gate) {
    float p0[4], p1[4];
#pragma unroll
    for (int w = 0; w < 4; ++w) {
      float A  = ang[gate * 4 + w] + cst[CST_TH + gate * 4 + w];
      float cA = __cosf(A);
      p0[w] = 0.5f + 0.5f * cA;
      p1[w] = 0.5f - 0.5f * cA;
    }
    float e0 = 0.f, e1 = 0.f, e2 = 0.f, e3 = 0.f;
#pragma unroll
    for (int i = 0; i < 16; ++i) {
      float pr = ((i & 8) ? p1[0] : p0[0]) * ((i & 4) ? p1[1] : p0[1]) *
                 ((i & 2) ? p1[2] : p0[2]) * ((i & 1) ? p1[3] : p0[3]);
      int b0 = (i >> 3) & 1, b1 = (i >> 2) & 1, b2 = (i >> 1) & 1, b3 = i & 1;
      b1 ^= b0; b2 ^= b1; b3 ^= b2; b0 ^= b3;   // CNOT(0,1)(1,2)(2,3)(3,0)
      e0 += b0 ? -pr : pr;
      e1 += b1 ? -pr : pr;
      e2 += b2 ? -pr : pr;
      e3 += b3 ? -pr : pr;
    }
    gates[gate * 4 + 0] = e0;
    gates[gate * 4 + 1] = e1;
    gates[gate * 4 + 2] = e2;
    gates[gate * 4 + 3] = e3;
  }

  float reg = cst[CST_BR];
#pragma unroll
  for (int w = 0; w < 4; ++w) {
    float fg = sigmoidf_(gates[w]);
    float ig = sigmoidf_(gates[4 + w]);
    float gg = tanhf(gates[8 + w]);
    float og = sigmoidf_(gates[12 + w]);
    c[w] = fg * c[w] + ig * gg;
    h[w] = og * tanhf(c[w]);
    stacked[((size_t)t * BB + b) * 4 + w] = h[w];
    hbuf[b * 4 + w] = h[w];
    cbuf[b * 4 + w] = c[w];
    reg += h[w] * cst[CST_WR + w];
  }
  regress[(size_t)t * BB + b] = reg;
}

extern "C" void kernel_launch(void* const* d_in, const int* in_sizes, int n_in,
                              void* d_out, int out_size, void* d_ws, size_t ws_size,
                              hipStream_t stream) {
  const float* X   = (const float*)d_in[0];
  const float* Wf  = (const float*)d_in[1];
  const float* bf  = (const float*)d_in[2];
  const float* Wi  = (const float*)d_in[3];
  const float* bi  = (const float*)d_in[4];
  const float* Wu  = (const float*)d_in[5];
  const float* bu  = (const float*)d_in[6];
  const float* Wo  = (const float*)d_in[7];
  const float* bo  = (const float*)d_in[8];
  const float* thf = (const float*)d_in[9];
  const float* thi = (const float*)d_in[10];
  const float* thu = (const float*)d_in[11];
  const float* tho = (const float*)d_in[12];
  const float* Wr  = (const float*)d_in[13];
  const float* br  = (const float*)d_in[14];
  float* out = (float*)d_out;

  float* cst = (float*)d_ws;                       // first 4KB: repacked constants
  float* pre = (float*)((char*)d_ws + 4096);       // T*B*16 f32 = 128 MB
  size_t need = 4096 + (size_t)TT * BB * 16 * sizeof(float);
  int usePre = (ws_size >= need) ? 1 : 0;

  prep_kernel<<<1, 256, 0, stream>>>(Wf, bf, Wi, bi, Wu, bu, Wo, bo,
                                     thf, thi, thu, tho, Wr, br, cst);
  init_state<<<(2 * BB * 4 + 255) / 256, 256, 0, stream>>>(out);

  if (usePre) {
    int numRows = TT * BB;                         // 2,097,152 rows, 131072 tiles
    int waves   = numRows / 16;
    gemm_pre_kernel<<<waves / 4, 128, 0, stream>>>(X, cst, pre, numRows);
  }

  const float* src = usePre ? (const float*)pre : X;
  for (int t = 0; t < TT; ++t) {
    step_kernel<<<BB / 256, 256, 0, stream>>>(src, cst, out, t, usePre);
  }
}